// MaskedEmbedding_11819749999085
// MI455X (gfx1250) — compile-verified
//
#include <hip/hip_runtime.h>
#include <stdint.h>

// MaskedEmbedding: out[tok, :] = (mask_real[x[tok], :] > 0.01f) ? weight[x[tok], :] : 0
// V=50257, D=768, tokens = B*S = 16384. Output f32 [16384, 768].
//
// Memory-bound gather (~150 MB @ 23.3 TB/s ~ 6.5 us floor). No matmul -> no WMMA.
// CDNA5 path used: async global->LDS copy engine (ASYNCcnt) to double-buffer the
// gathered rows per wave, hiding random-row latency without cross-wave barriers.

#define THRESHOLD 0.01f
#define DIM 768
#define VEC 4
#define THREADS (DIM / VEC)      // 192 threads = 6 waves (wave32)
#define TOK_PER_BLOCK 8

typedef float f4 __attribute__((ext_vector_type(4)));

__device__ __forceinline__ uint32_t lds_offset_of(const void* p) {
    // Generic pointers to LDS carry the wave-relative LDS byte offset in the
    // low 32 bits (CDNA5 ISA 10.2).
    return (uint32_t)(uint64_t)p;
}

__device__ __forceinline__ void async_copy_b128(uint32_t lds_byte_addr, uint64_t global_addr) {
    // GLOBAL_LOAD_ASYNC_TO_LDS_B128, GV mode (64-bit vaddr, saddr=off).
    // Per active lane: 16 bytes  memory -> LDS.  Tracked by ASYNCcnt.
    asm volatile("global_load_async_to_lds_b128 %0, %1, off"
                 :
                 : "v"(lds_byte_addr), "v"(global_addr)
                 : "memory");
}

__global__ __launch_bounds__(THREADS) void masked_embed_kernel(
    const int* __restrict__ x,
    const float* __restrict__ mask_real,
    const float* __restrict__ weight,
    float* __restrict__ out,
    int n_tokens)
{
    __shared__ __align__(16) float sm_mask[2][DIM];
    __shared__ __align__(16) float sm_wt[2][DIM];

    const int      tid      = threadIdx.x;
    const int      base_tok = blockIdx.x * TOK_PER_BLOCK;
    const uint32_t byte_off = (uint32_t)tid * (VEC * sizeof(float));   // 16*tid

    // This lane's private 16B slot in each LDS buffer (written by the async
    // engine, read back only by this lane -> no workgroup barrier needed).
    const uint32_t lmask0 = lds_offset_of(&sm_mask[0][0]) + byte_off;
    const uint32_t lmask1 = lds_offset_of(&sm_mask[1][0]) + byte_off;
    const uint32_t lwt0   = lds_offset_of(&sm_wt[0][0])   + byte_off;
    const uint32_t lwt1   = lds_offset_of(&sm_wt[1][0])   + byte_off;

    auto stage = [&](int buf, int tok) {
        int t = tok < n_tokens ? tok : n_tokens - 1;   // clamp: harmless extra load
        int idx = x[t];                                // uniform -> s_load
        uint64_t gm = (uint64_t)(mask_real + (size_t)idx * DIM) + byte_off;
        uint64_t gw = (uint64_t)(weight    + (size_t)idx * DIM) + byte_off;
        async_copy_b128(buf ? lmask1 : lmask0, gm);
        async_copy_b128(buf ? lwt1   : lwt0,   gw);
    };

    stage(0, base_tok);

    #pragma unroll
    for (int t = 0; t < TOK_PER_BLOCK; ++t) {
        const int buf = t & 1;
        if (t + 1 < TOK_PER_BLOCK) {
            stage(buf ^ 1, base_tok + t + 1);
            // Async ops complete in order: <=2 outstanding means token t's
            // two copies have landed in LDS.
            asm volatile("s_wait_asynccnt 0x2" ::: "memory");
        } else {
            asm volatile("s_wait_asynccnt 0x0" ::: "memory");
        }

        // ds_load_b128 of exactly the bytes this lane staged.
        f4 m = *(const f4*)&sm_mask[buf][tid * VEC];
        f4 w = *(const f4*)&sm_wt[buf][tid * VEC];

        f4 r;
        r.x = (m.x > THRESHOLD) ? w.x : 0.0f;
        r.y = (m.y > THRESHOLD) ? w.y : 0.0f;
        r.z = (m.z > THRESHOLD) ? w.z : 0.0f;
        r.w = (m.w > THRESHOLD) ? w.w : 0.0f;

        const int tok = base_tok + t;
        if (tok < n_tokens) {
            // Streaming 48 MB output: non-temporal so it doesn't evict gather
            // rows from L2 (global_store_b128 with NT hint).
            f4* dst = (f4*)(out + (size_t)tok * DIM) + tid;
            __builtin_nontemporal_store(r, dst);
        }
    }
}

extern "C" void kernel_launch(void* const* d_in, const int* in_sizes, int n_in,
                              void* d_out, int out_size, void* d_ws, size_t ws_size,
                              hipStream_t stream) {
    const int*   x         = (const int*)d_in[0];     // [B,S] int32
    const float* mask_real = (const float*)d_in[1];   // [V,D] f32
    const float* weight    = (const float*)d_in[2];   // [V,D] f32
    float*       out       = (float*)d_out;           // [B,S,D] f32

    const int n_tokens = in_sizes[0];                 // B*S = 16384
    const int blocks   = (n_tokens + TOK_PER_BLOCK - 1) / TOK_PER_BLOCK;

    masked_embed_kernel<<<blocks, THREADS, 0, stream>>>(x, mask_real, weight, out, n_tokens);
}